// energy_model_59227599011950
// MI455X (gfx1250) — compile-verified
//
#include <hip/hip_runtime.h>
#include <math.h>

#define N_NODES 8192
#define N_EDGES 131072

typedef __attribute__((ext_vector_type(2))) float v2f;
typedef __attribute__((ext_vector_type(8))) float v8f;
typedef int v4i __attribute__((vector_size(16)));

#if defined(__has_builtin)
#if __has_builtin(__builtin_amdgcn_global_load_async_to_lds_b128)
#define USE_ASYNC_LDS 1
#endif
#endif

__device__ __forceinline__ float siluf(float x) { return x / (1.f + __expf(-x)); }
__device__ __forceinline__ float sigmf(float x) { return 1.f / (1.f + __expf(-x)); }

__device__ __forceinline__ void atomicMaxF32(float* a, float v) {
  if (v >= 0.f) atomicMax((int*)a, __float_as_int(v));
  else          atomicMin((unsigned int*)a, __float_as_uint(v));
}

__device__ __forceinline__ float wave_sum(float x) {
#pragma unroll
  for (int off = 16; off > 0; off >>= 1) x += __shfl_xor(x, off, 32);
  return x;
}

__device__ __forceinline__ void wait_async_lds() {
#ifdef USE_ASYNC_LDS
#if __has_builtin(__builtin_amdgcn_s_wait_asynccnt)
  __builtin_amdgcn_s_wait_asynccnt(0);
#else
  asm volatile("s_wait_asynccnt 0" ::: "memory");
#endif
#endif
}

// ---------------------------------------------------------------------------
// Generic WMMA f32 GEMM:  C[M,N] = act(A[M,K] @ W[K,N] + bias)   (opt. C +=)
// BM=128 BN=32 BK=32, 256 threads = 8 waves; wave w owns rows w*16..w*16+15,
// two 16x16 WMMA tiles covering the 32-col block.
// A tile:  LDS [128][36] (rows 16B-aligned; conflict-free b64 fragment reads;
//          filled by GLOBAL_LOAD_ASYNC_TO_LDS_B128 when available).
// W tile:  LDS packed in fragment order: for k-step kk (of 4) and half h,
//          base = (kk/4)*192 + h*96 words, element (n,p) at base + 2n + p
//          holding W[kk+2h+p][n]  -> each b-fragment is ONE ds_load_b64,
//          halves 96 words apart -> disjoint bank sets, no conflicts.
// ---------------------------------------------------------------------------
template <int ACT, bool ADD, bool BIAS>
__global__ __launch_bounds__(256) void gemm_kernel(const float* __restrict__ A,
                                                   const float* __restrict__ W,
                                                   const float* __restrict__ bias,
                                                   float* __restrict__ C,
                                                   int M, int K, int N) {
  __shared__ float lA[128][36];
  __shared__ float lWp[8 * 192];
  const int tid  = threadIdx.x;
  const int wave = tid >> 5;
  const int lane = tid & 31;
  const int m0   = blockIdx.x * 128;
  const int n0   = blockIdx.y * 32;
  const int mr   = lane & 15;
  const int hi   = lane >> 4;
  const int koff = hi * 2;
  const int mrow = wave * 16 + mr;

  v8f c0 = {}, c1 = {};
  for (int k0 = 0; k0 < K; k0 += 32) {
    __syncthreads();
    {
      // ---- A tile: 128 x 32 floats ----
      int r  = tid >> 3;             // 0..31
      int cq = (tid & 7) * 4;        // col group
#ifdef USE_ASYNC_LDS
#pragma unroll
      for (int p = 0; p < 4; ++p) {
        int row = p * 32 + r;
        __builtin_amdgcn_global_load_async_to_lds_b128(
            (__attribute__((address_space(1))) v4i*)(A + (size_t)(m0 + row) * K + k0 + cq),
            (__attribute__((address_space(3))) v4i*)&lA[row][cq], 0, 0);
      }
#else
#pragma unroll
      for (int p = 0; p < 4; ++p) {
        int row = p * 32 + r;
        float4 val = *(const float4*)(A + (size_t)(m0 + row) * K + k0 + cq);
        lA[row][cq + 0] = val.x; lA[row][cq + 1] = val.y;
        lA[row][cq + 2] = val.z; lA[row][cq + 3] = val.w;
        if (k0 + 32 < K)
          __builtin_prefetch(A + (size_t)(m0 + row) * K + k0 + 32 + cq, 0, 3);
      }
#endif
      // ---- W tile: 32 x 32 floats, packed in fragment order ----
      int p  = tid >> 4;             // 0..15 -> k pair base k=2p
      int nb = (tid & 15) * 2;       // 0,2,..30
      float2 g0 = *(const float2*)(W + (size_t)(k0 + 2 * p + 0) * N + n0 + nb);
      float2 g1 = *(const float2*)(W + (size_t)(k0 + 2 * p + 1) * N + n0 + nb);
      float* wdst = lWp + (p >> 1) * 192 + (p & 1) * 96 + nb * 2;
      float4 pk; pk.x = g0.x; pk.y = g1.x; pk.z = g0.y; pk.w = g1.y;
      *(float4*)wdst = pk;
    }
    wait_async_lds();
    __syncthreads();
#pragma unroll
    for (int kk = 0; kk < 32; kk += 4) {
      v2f a, b0, b1;
      a = *(const v2f*)&lA[mrow][kk + koff];
      const float* wp = lWp + (kk >> 2) * 192 + hi * 96;
      b0 = *(const v2f*)(wp + 2 * mr);
      b1 = *(const v2f*)(wp + 2 * (16 + mr));
      c0 = __builtin_amdgcn_wmma_f32_16x16x4_f32(false, a, false, b0, (short)0, c0, false, false);
      c1 = __builtin_amdgcn_wmma_f32_16x16x4_f32(false, a, false, b1, (short)0, c1, false, false);
    }
  }
  float bv0 = BIAS ? bias[n0 + mr]      : 0.f;
  float bv1 = BIAS ? bias[n0 + 16 + mr] : 0.f;
#pragma unroll
  for (int i = 0; i < 8; ++i) {
    int row  = m0 + wave * 16 + i + 8 * hi;
    float x0 = c0[i] + bv0;
    float x1 = c1[i] + bv1;
    if (ACT == 1) { x0 = siluf(x0); x1 = siluf(x1); }
    else if (ACT == 2) { x0 = sigmf(x0); x1 = sigmf(x1); }
    size_t o0 = (size_t)row * N + n0 + mr;
    size_t o1 = o0 + 16;
    if (ADD) { C[o0] += x0; C[o1] += x1; } else { C[o0] = x0; C[o1] = x1; }
  }
}

template <int ACT, bool ADD, bool BIAS>
static inline void launch_gemm(const float* A, const float* W, const float* b, float* C,
                               int M, int K, int N, hipStream_t s) {
  dim3 grid(M / 128, N / 32);
  gemm_kernel<ACT, ADD, BIAS><<<grid, 256, 0, s>>>(A, W, b, C, M, K, N);
}

// ---------------------------------------------------------------------------
__global__ void fill_kernel(float* p, size_t n, float v) {
  size_t t = (size_t)blockIdx.x * blockDim.x + threadIdx.x;
  if (t < n) p[t] = v;
}
static inline void fill(float* p, size_t n, float v, hipStream_t s) {
  fill_kernel<<<(unsigned)((n + 255) / 256), 256, 0, s>>>(p, n, v);
}

// one wave per edge: vec, r, sh(8), rbf(128)
__global__ __launch_bounds__(256) void edge_geom_kernel(const float* __restrict__ pos,
                                                        const int* __restrict__ esrc,
                                                        const int* __restrict__ edst,
                                                        float* __restrict__ sh,
                                                        float* __restrict__ rb) {
  int e    = (int)(((size_t)blockIdx.x * blockDim.x + threadIdx.x) >> 5);
  int lane = threadIdx.x & 31;
  if (e >= N_EDGES) return;
  int s = esrc[e], d = edst[e];
  float vx = pos[s * 3 + 0] - pos[d * 3 + 0];
  float vy = pos[s * 3 + 1] - pos[d * 3 + 1];
  float vz = pos[s * 3 + 2] - pos[d * 3 + 2];
  float r  = sqrtf(vx * vx + vy * vy + vz * vz);
  float iv = 1.f / (r + 1e-12f);
  float x = vx * iv, y = vy * iv, z = vz * iv;
#pragma unroll
  for (int j = 0; j < 4; ++j) {
    int i    = lane + 32 * j;
    float c  = (5.0f / 127.0f) * (float)i;
    float dd = (r - c) * (128.0f / 5.0f);
    rb[(size_t)e * 128 + i] = __expf(-0.5f * dd * dd);
  }
  if (lane == 0) {
    const float c1 = 1.7320508075688772f;   // sqrt(3)
    const float ca = 3.872983346207417f;    // sqrt(15)
    const float cb = 1.118033988749895f;    // sqrt(5)/2
    const float cc = 1.9364916731037085f;   // sqrt(15)/2
    float* o = sh + (size_t)e * 8;
    o[0] = c1 * y;  o[1] = c1 * z;  o[2] = c1 * x;
    o[3] = ca * x * y;  o[4] = ca * y * z;
    o[5] = cb * (2.f * z * z - x * x - y * y);
    o[6] = ca * x * z;  o[7] = cc * (x * x - y * y);
  }
}

__global__ void node_embed_kernel(const int* __restrict__ atom,
                                  const float* __restrict__ Wemb,
                                  float* __restrict__ s) {
  int t = blockIdx.x * blockDim.x + threadIdx.x;
  if (t >= N_NODES * 128) return;
  int n = t >> 7, c = t & 127;
  s[t] = Wemb[atom[n] * 128 + c];
}

// one wave per edge: scatter deg-radial into s, v, q
__global__ __launch_bounds__(256) void scatter_deg_kernel(const float* __restrict__ wed,
                                                          const float* __restrict__ sh,
                                                          const int* __restrict__ edst,
                                                          float* __restrict__ sfeat,
                                                          float* __restrict__ v,
                                                          float* __restrict__ q) {
  int e    = (int)(((size_t)blockIdx.x * blockDim.x + threadIdx.x) >> 5);
  int lane = threadIdx.x & 31;
  if (e >= N_EDGES) return;
  int d = edst[e];
  const float sc = 0.044721359549995794f;  // 1/sqrt(500)
  const float* we = wed + (size_t)e * 224;
  float s1[3], s2[5];
#pragma unroll
  for (int k = 0; k < 3; ++k) s1[k] = sh[(size_t)e * 8 + k];
#pragma unroll
  for (int k = 0; k < 5; ++k) s2[k] = sh[(size_t)e * 8 + 3 + k];
#pragma unroll
  for (int j = 0; j < 4; ++j) {
    int ch = lane + 32 * j;
    atomicAdd(&sfeat[(size_t)d * 128 + ch], we[ch] * sc);
  }
#pragma unroll
  for (int ci = 0; ci < 2; ++ci) {
    int c = lane + 32 * ci;
    float wv = we[128 + c] * sc;
#pragma unroll
    for (int dd = 0; dd < 3; ++dd)
      atomicAdd(&v[((size_t)d * 64 + c) * 3 + dd], wv * s1[dd]);
  }
  {
    float wq = we[192 + lane] * sc;
#pragma unroll
    for (int dd = 0; dd < 5; ++dd)
      atomicAdd(&q[((size_t)d * 32 + lane) * 5 + dd], wq * s2[dd]);
  }
}

// one wave per node: equivariant layernorm -> hs, hv, hq
__global__ __launch_bounds__(256) void eqln_kernel(const float* __restrict__ s,
                                                   const float* __restrict__ v,
                                                   const float* __restrict__ q,
                                                   const float* __restrict__ g,
                                                   const float* __restrict__ b,
                                                   const float* __restrict__ gv,
                                                   const float* __restrict__ gq,
                                                   float* __restrict__ hs,
                                                   float* __restrict__ hv,
                                                   float* __restrict__ hq) {
  int n    = (int)(((size_t)blockIdx.x * blockDim.x + threadIdx.x) >> 5);
  int lane = threadIdx.x & 31;
  if (n >= N_NODES) return;
  float sv[4], sum = 0.f, sum2 = 0.f;
#pragma unroll
  for (int j = 0; j < 4; ++j) {
    sv[j] = s[(size_t)n * 128 + lane + 32 * j];
    sum += sv[j]; sum2 += sv[j] * sv[j];
  }
  float vv[6], vsum = 0.f;
#pragma unroll
  for (int ci = 0; ci < 2; ++ci)
#pragma unroll
    for (int dd = 0; dd < 3; ++dd) {
      float x = v[((size_t)n * 64 + lane + 32 * ci) * 3 + dd];
      vv[ci * 3 + dd] = x; vsum += x * x;
    }
  float qv[5], qsum = 0.f;
#pragma unroll
  for (int dd = 0; dd < 5; ++dd) {
    float x = q[((size_t)n * 32 + lane) * 5 + dd];
    qv[dd] = x; qsum += x * x;
  }
  sum = wave_sum(sum); sum2 = wave_sum(sum2);
  vsum = wave_sum(vsum); qsum = wave_sum(qsum);
  float mu  = sum * (1.f / 128.f);
  float var = sum2 * (1.f / 128.f) - mu * mu;
  float rs  = rsqrtf(var + 1e-5f);
  float inv_nv = 1.f / sqrtf(vsum * (1.f / 64.f) + 1e-5f);
  float inv_nq = 1.f / sqrtf(qsum * (1.f / 32.f) + 1e-5f);
#pragma unroll
  for (int j = 0; j < 4; ++j) {
    int ch = lane + 32 * j;
    hs[(size_t)n * 128 + ch] = (sv[j] - mu) * rs * g[ch] + b[ch];
  }
#pragma unroll
  for (int ci = 0; ci < 2; ++ci) {
    int c = lane + 32 * ci;
    float gn = gv[c] * inv_nv;
#pragma unroll
    for (int dd = 0; dd < 3; ++dd)
      hv[((size_t)n * 64 + c) * 3 + dd] = vv[ci * 3 + dd] * gn;
  }
  {
    float gn = gq[lane] * inv_nq;
#pragma unroll
    for (int dd = 0; dd < 5; ++dd)
      hq[((size_t)n * 32 + lane) * 5 + dd] = qv[dd] * gn;
  }
}

// one wave per edge: alpha[e,h] = sum_c silu(ws*hs[src]) * Wa ; atomic max per dst
__global__ __launch_bounds__(256) void alpha_kernel(const float* __restrict__ wed,
                                                    const float* __restrict__ hs,
                                                    const int* __restrict__ esrc,
                                                    const int* __restrict__ edst,
                                                    const float* __restrict__ Wa,
                                                    float* __restrict__ alpha,
                                                    float* __restrict__ nmax) {
  int e    = (int)(((size_t)blockIdx.x * blockDim.x + threadIdx.x) >> 5);
  int lane = threadIdx.x & 31;
  if (e >= N_EDGES) return;
  int sN = esrc[e], dN = edst[e];
  float part[4];
#pragma unroll
  for (int h = 0; h < 4; ++h) {
    int ch   = h * 32 + lane;
    float ms = wed[(size_t)e * 224 + ch] * hs[(size_t)sN * 128 + ch];
    part[h]  = siluf(ms) * Wa[ch];
  }
#pragma unroll
  for (int off = 16; off > 0; off >>= 1)
#pragma unroll
    for (int h = 0; h < 4; ++h) part[h] += __shfl_xor(part[h], off, 32);
  if (lane == 0) {
#pragma unroll
    for (int h = 0; h < 4; ++h) {
      alpha[(size_t)e * 4 + h] = part[h];
      atomicMaxF32(&nmax[(size_t)dN * 4 + h], part[h]);
    }
  }
}

// thread per (e,h): ex = exp(a - max); atomic sum per dst
__global__ void softmax_num_kernel(const float* __restrict__ alpha,
                                   const int* __restrict__ edst,
                                   const float* __restrict__ nmax,
                                   float* __restrict__ ex,
                                   float* __restrict__ nsum) {
  int t = blockIdx.x * blockDim.x + threadIdx.x;
  if (t >= N_EDGES * 4) return;
  int e = t >> 2, h = t & 3;
  int d = edst[e];
  float vv = __expf(alpha[t] - nmax[(size_t)d * 4 + h]);
  ex[t] = vv;
  atomicAdd(&nsum[(size_t)d * 4 + h], vv);
}

// one wave per edge: recompute messages, scatter attn-weighted sums
__global__ __launch_bounds__(256) void scatter_attn_kernel(const float* __restrict__ wed,
                                                           const float* __restrict__ sh,
                                                           const float* __restrict__ hs,
                                                           const float* __restrict__ hv,
                                                           const float* __restrict__ hq,
                                                           const int* __restrict__ esrc,
                                                           const int* __restrict__ edst,
                                                           const float* __restrict__ ex,
                                                           const float* __restrict__ nsum,
                                                           float* __restrict__ a_s,
                                                           float* __restrict__ a_v,
                                                           float* __restrict__ a_q) {
  int e    = (int)(((size_t)blockIdx.x * blockDim.x + threadIdx.x) >> 5);
  int lane = threadIdx.x & 31;
  if (e >= N_EDGES) return;
  int sN = esrc[e], dN = edst[e];
  float attn[4];
#pragma unroll
  for (int h = 0; h < 4; ++h)
    attn[h] = ex[(size_t)e * 4 + h] / (nsum[(size_t)dN * 4 + h] + 1e-9f);
  float s1[3], s2[5];
#pragma unroll
  for (int k = 0; k < 3; ++k) s1[k] = sh[(size_t)e * 8 + k];
#pragma unroll
  for (int k = 0; k < 5; ++k) s2[k] = sh[(size_t)e * 8 + 3 + k];
  float gs[4];
#pragma unroll
  for (int h = 0; h < 4; ++h) {
    int ch  = h * 32 + lane;
    gs[h]   = hs[(size_t)sN * 128 + ch];
    float m = wed[(size_t)e * 224 + ch] * gs[h];
    atomicAdd(&a_s[(size_t)dN * 128 + ch], m * attn[h]);
  }
#pragma unroll
  for (int ci = 0; ci < 2; ++ci) {
    int c    = lane + 32 * ci;
    float wv = wed[(size_t)e * 224 + 128 + c];
    float gg = gs[ci];
    float at = attn[c >> 4];
#pragma unroll
    for (int dd = 0; dd < 3; ++dd) {
      float m = wv * (hv[((size_t)sN * 64 + c) * 3 + dd] + gg * s1[dd]);
      atomicAdd(&a_v[((size_t)dN * 64 + c) * 3 + dd], m * at);
    }
  }
  {
    int c    = lane;
    float wq = wed[(size_t)e * 224 + 192 + c];
    float gg = gs[0];
    float at = attn[c >> 3];
#pragma unroll
    for (int dd = 0; dd < 5; ++dd) {
      float m = wq * (hq[((size_t)sN * 32 + c) * 5 + dd] + gg * s2[dd]);
      atomicAdd(&a_q[((size_t)dN * 32 + c) * 5 + dd], m * at);
    }
  }
}

// 8 nodes per block: v += a_v @ Wp_v (per d), q += a_q @ Wp_q (per d)
__global__ __launch_bounds__(256) void proj_vq_kernel(const float* __restrict__ a_v,
                                                      const float* __restrict__ a_q,
                                                      const float* __restrict__ Wv,
                                                      const float* __restrict__ Wq,
                                                      float* __restrict__ v,
                                                      float* __restrict__ q) {
  __shared__ float sWv[64 * 64];
  __shared__ float sWq[32 * 32];
  __shared__ float sav[8][192];
  __shared__ float saq[8][160];
  int tid = threadIdx.x;
  for (int i = tid; i < 64 * 64; i += 256) sWv[i] = Wv[i];
  for (int i = tid; i < 32 * 32; i += 256) sWq[i] = Wq[i];
  int nb = blockIdx.x * 8;
  for (int i = tid; i < 8 * 192; i += 256) (&sav[0][0])[i] = a_v[(size_t)nb * 192 + i];
  for (int i = tid; i < 8 * 160; i += 256) (&saq[0][0])[i] = a_q[(size_t)nb * 160 + i];
  __syncthreads();
  int w = tid >> 5, lane = tid & 31;
  int n = nb + w;
#pragma unroll
  for (int mi = 0; mi < 2; ++mi) {
    int m = lane + 32 * mi;
    float a0 = 0.f, a1 = 0.f, a2 = 0.f;
    for (int c = 0; c < 64; ++c) {
      float wv = sWv[c * 64 + m];
      a0 += sav[w][c * 3 + 0] * wv;
      a1 += sav[w][c * 3 + 1] * wv;
      a2 += sav[w][c * 3 + 2] * wv;
    }
    v[((size_t)n * 64 + m) * 3 + 0] += a0;
    v[((size_t)n * 64 + m) * 3 + 1] += a1;
    v[((size_t)n * 64 + m) * 3 + 2] += a2;
  }
  {
    float acc[5] = {0.f, 0.f, 0.f, 0.f, 0.f};
    for (int c = 0; c < 32; ++c) {
      float wq = sWq[c * 32 + lane];
#pragma unroll
      for (int dd = 0; dd < 5; ++dd) acc[dd] += saq[w][c * 5 + dd] * wq;
    }
#pragma unroll
    for (int dd = 0; dd < 5; ++dd) q[((size_t)n * 32 + lane) * 5 + dd] += acc[dd];
  }
}

// 8 nodes per block: FFN equivariant branch:
// m_v = (hv@W1v)*gate[:, :64] ; v += m_v@W2v ; same for q with gate[:, 64:96]
__global__ __launch_bounds__(256) void ffn_vq_kernel(const float* __restrict__ hv,
                                                     const float* __restrict__ hq,
                                                     const float* __restrict__ gate,
                                                     const float* __restrict__ W1v,
                                                     const float* __restrict__ W1q,
                                                     const float* __restrict__ W2v,
                                                     const float* __restrict__ W2q,
                                                     float* __restrict__ v,
                                                     float* __restrict__ q) {
  __shared__ float sW1v[64 * 64];
  __shared__ float sW2v[64 * 64];
  __shared__ float sW1q[32 * 32];
  __shared__ float sW2q[32 * 32];
  __shared__ float shv[8][192];
  __shared__ float shq[8][160];
  __shared__ float stv[8][192];
  __shared__ float stq[8][160];
  int tid = threadIdx.x;
  for (int i = tid; i < 64 * 64; i += 256) { sW1v[i] = W1v[i]; sW2v[i] = W2v[i]; }
  for (int i = tid; i < 32 * 32; i += 256) { sW1q[i] = W1q[i]; sW2q[i] = W2q[i]; }
  int nb = blockIdx.x * 8;
  for (int i = tid; i < 8 * 192; i += 256) (&shv[0][0])[i] = hv[(size_t)nb * 192 + i];
  for (int i = tid; i < 8 * 160; i += 256) (&shq[0][0])[i] = hq[(size_t)nb * 160 + i];
  __syncthreads();
  int w = tid >> 5, lane = tid & 31;
  int n = nb + w;
#pragma unroll
  for (int mi = 0; mi < 2; ++mi) {
    int m = lane + 32 * mi;
    float a0 = 0.f, a1 = 0.f, a2 = 0.f;
    for (int c = 0; c < 64; ++c) {
      float wv = sW1v[c * 64 + m];
      a0 += shv[w][c * 3 + 0] * wv;
      a1 += shv[w][c * 3 + 1] * wv;
      a2 += shv[w][c * 3 + 2] * wv;
    }
    float gt = gate[(size_t)n * 96 + m];
    stv[w][m * 3 + 0] = a0 * gt;
    stv[w][m * 3 + 1] = a1 * gt;
    stv[w][m * 3 + 2] = a2 * gt;
  }
  {
    float acc[5] = {0.f, 0.f, 0.f, 0.f, 0.f};
    for (int c = 0; c < 32; ++c) {
      float wq = sW1q[c * 32 + lane];
#pragma unroll
      for (int dd = 0; dd < 5; ++dd) acc[dd] += shq[w][c * 5 + dd] * wq;
    }
    float gt = gate[(size_t)n * 96 + 64 + lane];
#pragma unroll
    for (int dd = 0; dd < 5; ++dd) stq[w][lane * 5 + dd] = acc[dd] * gt;
  }
  __syncthreads();
#pragma unroll
  for (int mi = 0; mi < 2; ++mi) {
    int m = lane + 32 * mi;
    float a0 = 0.f, a1 = 0.f, a2 = 0.f;
    for (int c = 0; c < 64; ++c) {
      float wv = sW2v[c * 64 + m];
      a0 += stv[w][c * 3 + 0] * wv;
      a1 += stv[w][c * 3 + 1] * wv;
      a2 += stv[w][c * 3 + 2] * wv;
    }
    v[((size_t)n * 64 + m) * 3 + 0] += a0;
    v[((size_t)n * 64 + m) * 3 + 1] += a1;
    v[((size_t)n * 64 + m) * 3 + 2] += a2;
  }
  {
    float acc[5] = {0.f, 0.f, 0.f, 0.f, 0.f};
    for (int c = 0; c < 32; ++c) {
      float wq = sW2q[c * 32 + lane];
#pragma unroll
      for (int dd = 0; dd < 5; ++dd) acc[dd] += stq[w][c * 5 + dd] * wq;
    }
#pragma unroll
    for (int dd = 0; dd < 5; ++dd) q[((size_t)n * 32 + lane) * 5 + dd] += acc[dd];
  }
}

// wave per node: LN over 512 channels, in place
__global__ __launch_bounds__(256) void ln512_kernel(float* __restrict__ f,
                                                    const float* __restrict__ g,
                                                    const float* __restrict__ b) {
  int n    = (int)(((size_t)blockIdx.x * blockDim.x + threadIdx.x) >> 5);
  int lane = threadIdx.x & 31;
  if (n >= N_NODES) return;
  float x[16], sum = 0.f, sum2 = 0.f;
#pragma unroll
  for (int j = 0; j < 16; ++j) {
    x[j] = f[(size_t)n * 512 + lane + 32 * j];
    sum += x[j]; sum2 += x[j] * x[j];
  }
  sum = wave_sum(sum); sum2 = wave_sum(sum2);
  float mu  = sum * (1.f / 512.f);
  float var = sum2 * (1.f / 512.f) - mu * mu;
  float rs  = rsqrtf(var + 1e-5f);
#pragma unroll
  for (int j = 0; j < 16; ++j) {
    int ch = lane + 32 * j;
    f[(size_t)n * 512 + ch] = (x[j] - mu) * rs * g[ch] + b[ch];
  }
}

// ---------------------------------------------------------------------------
extern "C" void kernel_launch(void* const* d_in, const int* in_sizes, int n_in,
                              void* d_out, int out_size, void* d_ws, size_t ws_size,
                              hipStream_t stream) {
  (void)in_sizes; (void)n_in; (void)out_size; (void)ws_size;
  const size_t E = N_EDGES, Nn = N_NODES;
  float* ws = (float*)d_ws;
  size_t off = 0;
  float* RB   = ws + off; off += E * 128;
  float* SH   = ws + off; off += E * 8;
  float* H1   = ws + off; off += E * 64;
  float* H2   = ws + off; off += E * 64;
  float* WED  = ws + off; off += E * 224;
  float* S    = ws + off; off += Nn * 128;
  float* V    = ws + off; off += Nn * 192;
  float* Q    = ws + off; off += Nn * 160;
  float* HSb  = ws + off; off += Nn * 128;
  float* HVb  = ws + off; off += Nn * 192;
  float* HQb  = ws + off; off += Nn * 160;
  float* AS   = ws + off; off += Nn * 128;
  float* AV   = ws + off; off += Nn * 192;
  float* AQ   = ws + off; off += Nn * 160;
  float* ALP  = ws + off; off += E * 4;
  float* EXb  = ws + off; off += E * 4;
  float* NMX  = ws + off; off += Nn * 4;
  float* NSM  = ws + off; off += Nn * 4;
  float* Fb   = ws + off; off += Nn * 512;
  float* GATE = ws + off; off += Nn * 96;
  float* MSB  = ws + off; off += Nn * 128;
  float* HB   = ws + off; off += Nn * 512;

  const float* Wemb   = (const float*)d_in[8];
  const float* degW1  = (const float*)d_in[9];
  const float* degb1  = (const float*)d_in[10];
  const float* degW2  = (const float*)d_in[11];
  const float* degb2  = (const float*)d_in[12];
  const float* degW3  = (const float*)d_in[13];
  const float* degb3  = (const float*)d_in[14];
  const float* attW1  = (const float*)d_in[15];
  const float* attb1  = (const float*)d_in[16];
  const float* attW2  = (const float*)d_in[17];
  const float* attb2  = (const float*)d_in[18];
  const float* attW3  = (const float*)d_in[19];
  const float* attb3  = (const float*)d_in[20];
  const float* Wa     = (const float*)d_in[21];
  const float* WpS    = (const float*)d_in[22];
  const float* WpV    = (const float*)d_in[23];
  const float* WpQ    = (const float*)d_in[24];
  const float* ln1g   = (const float*)d_in[25];
  const float* ln1b   = (const float*)d_in[26];
  const float* ln1gv  = (const float*)d_in[27];
  const float* ln1gq  = (const float*)d_in[28];
  const float* ln2g   = (const float*)d_in[29];
  const float* ln2b   = (const float*)d_in[30];
  const float* ln2gv  = (const float*)d_in[31];
  const float* ln2gq  = (const float*)d_in[32];
  const float* fW1s   = (const float*)d_in[33];
  const float* fW1v   = (const float*)d_in[34];
  const float* fW1q   = (const float*)d_in[35];
  const float* fWg    = (const float*)d_in[36];
  const float* fW2s   = (const float*)d_in[37];
  const float* fW2v   = (const float*)d_in[38];
  const float* fW2q   = (const float*)d_in[39];
  const float* lnfg   = (const float*)d_in[40];
  const float* lnfb   = (const float*)d_in[41];
  const float* lnfgv  = (const float*)d_in[42];
  const float* lnfgq  = (const float*)d_in[43];
  const float* Wlast  = (const float*)d_in[44];
  const float* normg  = (const float*)d_in[45];
  const float* normb  = (const float*)d_in[46];
  const float* Wh1    = (const float*)d_in[47];
  const float* bh1    = (const float*)d_in[48];
  const float* Wh2    = (const float*)d_in[49];
  const float* bh2    = (const float*)d_in[50];

  const int EDGE_WBLK = N_EDGES / 8;   // wave-per-edge blocks (256 thr)
  const int NODE_WBLK = N_NODES / 8;   // wave-per-node blocks

  for (int g = 0; g < 2; ++g) {
    const float* pos = (const float*)d_in[g * 4 + 0];
    const int* atom  = (const int*)d_in[g * 4 + 1];
    const int* esrc  = (const int*)d_in[g * 4 + 2];
    const int* edst  = (const int*)d_in[g * 4 + 3];
    float* outp      = (float*)d_out + (size_t)g * Nn * 128;

    edge_geom_kernel<<<EDGE_WBLK, 256, 0, stream>>>(pos, esrc, edst, SH, RB);
    node_embed_kernel<<<(N_NODES * 128) / 256, 256, 0, stream>>>(atom, Wemb, S);
    fill(V, Nn * 192, 0.f, stream);
    fill(Q, Nn * 160, 0.f, stream);

    // degree-embedding radial MLP + scatter
    launch_gemm<1, false, true>(RB, degW1, degb1, H1, N_EDGES, 128, 64, stream);
    launch_gemm<1, false, true>(H1, degW2, degb2, H2, N_EDGES, 64, 64, stream);
    launch_gemm<0, false, true>(H2, degW3, degb3, WED, N_EDGES, 64, 224, stream);
    scatter_deg_kernel<<<EDGE_WBLK, 256, 0, stream>>>(WED, SH, edst, S, V, Q);

    for (int l = 0; l < 4; ++l) {
      eqln_kernel<<<NODE_WBLK, 256, 0, stream>>>(S, V, Q,
          ln1g + l * 128, ln1b + l * 128, ln1gv + l * 64, ln1gq + l * 32,
          HSb, HVb, HQb);
      launch_gemm<1, false, true>(RB, attW1 + l * 128 * 64, attb1 + l * 64, H1, N_EDGES, 128, 64, stream);
      launch_gemm<1, false, true>(H1, attW2 + l * 64 * 64, attb2 + l * 64, H2, N_EDGES, 64, 64, stream);
      launch_gemm<0, false, true>(H2, attW3 + l * 64 * 224, attb3 + l * 224, WED, N_EDGES, 64, 224, stream);

      fill(NMX, Nn * 4, -INFINITY, stream);
      alpha_kernel<<<EDGE_WBLK, 256, 0, stream>>>(WED, HSb, esrc, edst, Wa + l * 128, ALP, NMX);
      fill(NSM, Nn * 4, 0.f, stream);
      softmax_num_kernel<<<(N_EDGES * 4) / 256, 256, 0, stream>>>(ALP, edst, NMX, EXb, NSM);
      fill(AS, Nn * 128, 0.f, stream);
      fill(AV, Nn * 192, 0.f, stream);
      fill(AQ, Nn * 160, 0.f, stream);
      scatter_attn_kernel<<<EDGE_WBLK, 256, 0, stream>>>(WED, SH, HSb, HVb, HQb,
          esrc, edst, EXb, NSM, AS, AV, AQ);

      launch_gemm<0, true, false>(AS, WpS + l * 128 * 128, nullptr, S, N_NODES, 128, 128, stream);
      proj_vq_kernel<<<NODE_WBLK, 256, 0, stream>>>(AV, AQ, WpV + l * 64 * 64, WpQ + l * 32 * 32, V, Q);

      if (l < 3) {
        eqln_kernel<<<NODE_WBLK, 256, 0, stream>>>(S, V, Q,
            ln2g + l * 128, ln2b + l * 128, ln2gv + l * 64, ln2gq + l * 32,
            HSb, HVb, HQb);
        launch_gemm<1, false, false>(HSb, fW1s + l * 128 * 128, nullptr, MSB, N_NODES, 128, 128, stream);
        launch_gemm<2, false, false>(HSb, fWg + l * 128 * 96, nullptr, GATE, N_NODES, 128, 96, stream);
        ffn_vq_kernel<<<NODE_WBLK, 256, 0, stream>>>(HVb, HQb, GATE,
            fW1v + l * 64 * 64, fW1q + l * 32 * 32,
            fW2v + l * 64 * 64, fW2q + l * 32 * 32, V, Q);
        launch_gemm<0, true, false>(MSB, fW2s + l * 128 * 128, nullptr, S, N_NODES, 128, 128, stream);
      }
    }

    // head
    eqln_kernel<<<NODE_WBLK, 256, 0, stream>>>(S, V, Q, lnfg, lnfb, lnfgv, lnfgq, HSb, HVb, HQb);
    launch_gemm<0, false, false>(HSb, Wlast, nullptr, Fb, N_NODES, 128, 512, stream);
    ln512_kernel<<<NODE_WBLK, 256, 0, stream>>>(Fb, normg, normb);
    launch_gemm<1, false, true>(Fb, Wh1, bh1, HB, N_NODES, 512, 512, stream);
    launch_gemm<0, false, true>(HB, Wh2, bh2, outp, N_NODES, 512, 128, stream);
  }
}